// DeformationGrid_22909355557520
// MI455X (gfx1250) — compile-verified
//
#include <hip/hip_runtime.h>
#include <hip/hip_bf16.h>

typedef float f32x4 __attribute__((ext_vector_type(4)));

// 6 consecutive floats: both z-corners (k0, k0+1) of one (i,j) column, all 3 channels.
struct V6 { float v[6]; };

constexpr int G        = 128;
constexpr int CH       = 3;
constexpr int STRIDE_J = G * CH;       // 384 floats
constexpr int STRIDE_I = G * G * CH;   // 49152 floats

__device__ __forceinline__ int clampi(int v, int lo, int hi) {
    return v < lo ? lo : (v > hi ? hi : v);
}

// Trilinear interpolation of one point; r[0..2] = 3-channel result.
// Arithmetic follows the reference order exactly: lerp x, then y, then z.
__device__ __forceinline__ void trilerp_one(float cx, float cy, float cz,
                                            const float* __restrict__ g,
                                            float r[3]) {
    float x = cx * 127.0f;
    float y = cy * 127.0f;
    float z = cz * 127.0f;
    float fx = floorf(x), fy = floorf(y), fz = floorf(z);
    float wx = x - fx,    wy = y - fy,    wz = z - fz;
    int i0 = clampi((int)fx, 0, G - 2);
    int j0 = clampi((int)fy, 0, G - 2);
    int k0 = clampi((int)fz, 0, G - 2);

    const float* p00 = g + i0 * STRIDE_I + j0 * STRIDE_J + k0 * CH;
    const float* p01 = p00 + STRIDE_J;   // (i0, j1)
    const float* p10 = p00 + STRIDE_I;   // (i1, j0)
    const float* p11 = p10 + STRIDE_J;   // (i1, j1)

    // 4 chunk gathers of 24B each (z-pair fused) instead of 8x12B.
    V6 q00 = *(const V6*)p00;
    V6 q01 = *(const V6*)p01;
    V6 q10 = *(const V6*)p10;
    V6 q11 = *(const V6*)p11;

    float omx = 1.0f - wx, omy = 1.0f - wy, omz = 1.0f - wz;
#pragma unroll
    for (int c = 0; c < 3; ++c) {
        float c000 = q00.v[c], c001 = q00.v[c + 3];
        float c010 = q01.v[c], c011 = q01.v[c + 3];
        float c100 = q10.v[c], c101 = q10.v[c + 3];
        float c110 = q11.v[c], c111 = q11.v[c + 3];
        float c00 = c000 * omx + c100 * wx;
        float c01 = c001 * omx + c101 * wx;
        float c10 = c010 * omx + c110 * wx;
        float c11 = c011 * omx + c111 * wx;
        float c0  = c00 * omy + c10 * wy;
        float c1  = c01 * omy + c11 * wy;
        r[c]      = c0 * omz + c1 * wz;
    }
}

// Main kernel: 4 points per thread -> coords/out move as aligned b128 vectors,
// non-temporal (streamed once; keep the 24MiB grid resident in the 192MB L2).
__global__ __launch_bounds__(256)
void DeformationGrid_trilerp4(const float* __restrict__ coords,
                              const float* __restrict__ grid,
                              float* __restrict__ out,
                              int nquads) {
    const int stride = gridDim.x * blockDim.x;
    for (int q = blockIdx.x * blockDim.x + threadIdx.x; q < nquads; q += stride) {
        // gfx1250 global_prefetch_b8: warm next grid-stride tile of coords.
        // Speculative prefetch; OOB addresses are silently dropped.
        __builtin_prefetch(coords + (size_t)(q + stride) * 12, 0, 1);

        const f32x4* cin = (const f32x4*)(coords + (size_t)q * 12);
        f32x4 a = __builtin_nontemporal_load(cin + 0);
        f32x4 b = __builtin_nontemporal_load(cin + 1);
        f32x4 c = __builtin_nontemporal_load(cin + 2);

        float r0[3], r1[3], r2[3], r3[3];
        trilerp_one(a.x, a.y, a.z, grid, r0);
        trilerp_one(a.w, b.x, b.y, grid, r1);
        trilerp_one(b.z, b.w, c.x, grid, r2);
        trilerp_one(c.y, c.z, c.w, grid, r3);

        f32x4 o0 = { r0[0], r0[1], r0[2], r1[0] };
        f32x4 o1 = { r1[1], r1[2], r2[0], r2[1] };
        f32x4 o2 = { r2[2], r3[0], r3[1], r3[2] };

        f32x4* op = (f32x4*)(out + (size_t)q * 12);
        __builtin_nontemporal_store(o0, op + 0);
        __builtin_nontemporal_store(o1, op + 1);
        __builtin_nontemporal_store(o2, op + 2);
    }
}

// Generic tail for n % 4 != 0 (not hit with N = 4194304, but kept for safety).
__global__ void DeformationGrid_trilerp_tail(const float* __restrict__ coords,
                                             const float* __restrict__ grid,
                                             float* __restrict__ out,
                                             int start, int n) {
    int i = start + blockIdx.x * blockDim.x + threadIdx.x;
    if (i >= n) return;
    float r[3];
    trilerp_one(coords[(size_t)i * 3 + 0],
                coords[(size_t)i * 3 + 1],
                coords[(size_t)i * 3 + 2], grid, r);
    out[(size_t)i * 3 + 0] = r[0];
    out[(size_t)i * 3 + 1] = r[1];
    out[(size_t)i * 3 + 2] = r[2];
}

extern "C" void kernel_launch(void* const* d_in, const int* in_sizes, int n_in,
                              void* d_out, int out_size, void* d_ws, size_t ws_size,
                              hipStream_t stream) {
    const float* coords = (const float*)d_in[0];   // [N,3] f32
    const float* grid   = (const float*)d_in[1];   // [128,128,128,3] f32
    float*       out    = (float*)d_out;           // [N,3] f32

    const long n      = (long)in_sizes[0] / 3;
    const int  nquads = (int)(n / 4);
    const int  rem    = (int)(n - (long)nquads * 4);

    if (nquads > 0) {
        const int block   = 256;                       // 8 wave32 waves
        int       needed  = (nquads + block - 1) / block;
        int       nblocks = needed < 2048 ? needed : 2048;  // grid-stride so prefetch is live
        DeformationGrid_trilerp4<<<nblocks, block, 0, stream>>>(coords, grid, out, nquads);
    }
    if (rem > 0) {
        DeformationGrid_trilerp_tail<<<1, 64, 0, stream>>>(coords, grid, out,
                                                           nquads * 4, (int)n);
    }
}